// OptimalTransport_68813966016555
// MI455X (gfx1250) — compile-verified
//
#include <hip/hip_runtime.h>
#include <hip/hip_bf16.h>
#include <math.h>

// Problem constants (match reference)
#define NN 1024              // H*W
#define BB 8
#define CC 256
#define REG_INV 10.0f        // 1/REG
#define EPS_ 1e-6f
#define A_REAL (0.8f / 1024.0f)
#define A_VIRT 0.2f
#define UV_STRIDE 1040       // 1025 padded to multiple of 16 floats (16B align for float4)
#define NCHUNK 8             // split-K chunks for transposed matvec
#define CHUNK (NN / NCHUNK)  // 128 rows per chunk

typedef __attribute__((ext_vector_type(2))) float v2f;
typedef __attribute__((ext_vector_type(8))) float v8f;

// ---- workspace layout (float element offsets) ----
#define OFF_K      0u
#define OFF_COST0  (8u * 1024u * 1024u)            // 1M floats: cost matrix of batch 0
#define OFF_X2     (OFF_COST0 + 1024u * 1024u)
#define OFF_Y2     (OFF_X2 + 8u * 1024u)
#define OFF_U      (OFF_Y2 + 8u * 1024u)           // byte offset divisible by 16
#define OFF_V      (OFF_U + 8u * UV_STRIDE)
#define OFF_CS     (OFF_V + 8u * UV_STRIDE)        // column sums for final normalization
#define OFF_RADIX  (OFF_CS + 8u * 1024u)           // uint region: [0..255]=counts, [256]=prefix, [257]=krem
#define OFF_QRES   (OFF_RADIX + 258u)              // 2 floats: order statistics lo/hi
#define OFF_KV     (OFF_QRES + 2u)                 // 1 float: kv = max(exp(-virt/REG),1e-8)
#define OFF_PART   (OFF_KV + 6u)                   // 8B aligned; NCHUNK*B*N partial sums
// OFF_PART = 9478666 -> pad to 9478668? ensure /4: computed below statically as aligned
// (OFF_KV+6 = 9478666; 9478666*4 bytes is 8B aligned; partials accessed as scalar floats only)

// ---------------------------------------------------------------------------
// 1) squared norms: x2[b][n] = sum_c src[b][c][n]^2 (z=0), same for tgt (z=1)
__global__ void norms_kernel(const float* __restrict__ src,
                             const float* __restrict__ tgt,
                             float* __restrict__ ws) {
    const int b = blockIdx.y;
    const int n = blockIdx.x * 256 + threadIdx.x;
    const float* p = (blockIdx.z == 0 ? src : tgt) + (size_t)b * CC * NN + n;
    float s = 0.f;
    #pragma unroll 4
    for (int c = 0; c < CC; ++c) { float v = p[(size_t)c * NN]; s += v * v; }
    ws[(blockIdx.z == 0 ? OFF_X2 : OFF_Y2) + b * NN + n] = s;
}

// ---------------------------------------------------------------------------
// 2) cost GEMM via v_wmma_f32_16x16x4_f32, fused exp -> K; batch0 cost saved.
//    xy[n][m] = sum_c S[c][n] * T[c][m]   (S,T are [C, N] row-major)
//    Each wave: 16x64 output strip; block = 8 waves -> 16x512; grid (2,64,8).
__global__ __launch_bounds__(256)
void gemm_k_kernel(const float* __restrict__ src, const float* __restrict__ tgt,
                   float* __restrict__ ws) {
    const int b    = blockIdx.z;
    const int n0   = blockIdx.y * 16;
    const int wave = threadIdx.x >> 5;
    const int lane = threadIdx.x & 31;
    const int m0   = blockIdx.x * 512 + wave * 64;
    const int lm   = lane & 15;          // row/col within 16
    const int lk   = (lane >> 4) * 2;    // K-pair base per A/B fp32 layout

    const float* S = src + (size_t)b * CC * NN;
    const float* T = tgt + (size_t)b * CC * NN;

    v8f acc[4];
    #pragma unroll
    for (int t = 0; t < 4; ++t) acc[t] = (v8f){0.f,0.f,0.f,0.f,0.f,0.f,0.f,0.f};

    for (int c = 0; c < CC; c += 4) {
        const float* Sr0 = S + (size_t)(c + lk) * NN;
        const float* Sr1 = S + (size_t)(c + lk + 1) * NN;
        const float* Tr0 = T + (size_t)(c + lk) * NN;
        const float* Tr1 = T + (size_t)(c + lk + 1) * NN;
        v2f a;  a.x  = Sr0[n0 + lm];      a.y  = Sr1[n0 + lm];
        v2f b0; b0.x = Tr0[m0 + lm];      b0.y = Tr1[m0 + lm];
        v2f b1; b1.x = Tr0[m0 + 16 + lm]; b1.y = Tr1[m0 + 16 + lm];
        v2f b2; b2.x = Tr0[m0 + 32 + lm]; b2.y = Tr1[m0 + 32 + lm];
        v2f b3; b3.x = Tr0[m0 + 48 + lm]; b3.y = Tr1[m0 + 48 + lm];
        acc[0] = __builtin_amdgcn_wmma_f32_16x16x4_f32(false, a, false, b0, (short)0, acc[0], false, false);
        acc[1] = __builtin_amdgcn_wmma_f32_16x16x4_f32(false, a, false, b1, (short)0, acc[1], false, false);
        acc[2] = __builtin_amdgcn_wmma_f32_16x16x4_f32(false, a, false, b2, (short)0, acc[2], false, false);
        acc[3] = __builtin_amdgcn_wmma_f32_16x16x4_f32(false, a, false, b3, (short)0, acc[3], false, false);
    }

    const float* x2 = ws + OFF_X2 + b * NN;
    const float* y2 = ws + OFF_Y2 + b * NN;
    float* Kb    = ws + OFF_K + (size_t)b * NN * NN;
    float* cost0 = ws + OFF_COST0;

    float x2r[8];
    #pragma unroll
    for (int r = 0; r < 8; ++r) x2r[r] = x2[n0 + (lane >> 4) * 8 + r];

    #pragma unroll
    for (int t = 0; t < 4; ++t) {
        const int m = m0 + t * 16 + lm;
        const float ym = y2[m];
        #pragma unroll
        for (int r = 0; r < 8; ++r) {
            const int n = n0 + (lane >> 4) * 8 + r;
            float cost = fmaxf(x2r[r] + ym - 2.0f * acc[t][r], 0.0f);
            Kb[(size_t)n * NN + m] = fmaxf(expf(-cost * REG_INV), 1e-8f);
            if (b == 0) cost0[(size_t)n * NN + m] = cost;
        }
    }
}

// ---------------------------------------------------------------------------
// 3) exact radix-select of k-th smallest over batch-0 cost (bits monotone: all >= 0)
__global__ void radix_init_kernel(float* ws, unsigned k) {
    unsigned* R = (unsigned*)(ws + OFF_RADIX);
    if (threadIdx.x < 256) R[threadIdx.x] = 0u;
    if (threadIdx.x == 0) { R[256] = 0u; R[257] = k; }
}

__global__ __launch_bounds__(256)
void radix_count_kernel(float* ws, int shift, unsigned mask) {
    __shared__ unsigned h[256];
    for (int i = threadIdx.x; i < 256; i += 256) h[i] = 0u;
    __syncthreads();
    unsigned* R = (unsigned*)(ws + OFF_RADIX);
    const unsigned prefix = R[256];
    const unsigned* data = (const unsigned*)(ws + OFF_COST0);
    const unsigned stride = gridDim.x * blockDim.x;
    for (unsigned i = blockIdx.x * blockDim.x + threadIdx.x; i < 1048576u; i += stride) {
        unsigned bv = data[i];
        if ((bv & mask) == prefix) atomicAdd(&h[(bv >> shift) & 255u], 1u);
    }
    __syncthreads();
    for (int i = threadIdx.x; i < 256; i += 256)
        if (h[i]) atomicAdd(&R[i], h[i]);
}

__global__ void radix_scan_kernel(float* ws, int shift, int isLast, int sel) {
    if (threadIdx.x != 0) return;
    unsigned* R = (unsigned*)(ws + OFF_RADIX);
    unsigned krem = R[257], prefix = R[256];
    unsigned cum = 0u, d = 255u;
    for (unsigned i = 0; i < 256u; ++i) {
        unsigned c = R[i];
        if (krem < cum + c) { d = i; break; }
        cum += c;
    }
    prefix |= d << shift;
    R[256] = prefix;
    R[257] = krem - cum;
    for (int i = 0; i < 256; ++i) R[i] = 0u;
    if (isLast) ws[OFF_QRES + sel] = __uint_as_float(prefix);
}

__global__ void finalize_kv_kernel(float* ws) {
    if (threadIdx.x == 0) {
        float vlo = ws[OFF_QRES], vhi = ws[OFF_QRES + 1];
        float virt = vlo + 0.25f * (vhi - vlo);       // linear interp, frac = .25
        ws[OFF_KV] = fmaxf(expf(-virt * REG_INV), 1e-8f);
    }
}

__global__ void init_uv_kernel(float* ws) {
    const int b = blockIdx.x;
    for (int i = threadIdx.x; i <= NN; i += 256) {
        ws[OFF_U + b * UV_STRIDE + i] = 1.0f;
        ws[OFF_V + b * UV_STRIDE + i] = 1.0f;
    }
}

// ---------------------------------------------------------------------------
// 4a) Sinkhorn row half-step. Virtual row/col folded in via scalar kv.
//    u_i = a_i / (sum_j K_ij v_j + kv*v[N] + eps);  u_N = a_v/(kv*Σv + v[N] + eps)
__global__ __launch_bounds__(256)
void sink_rows_kernel(float* ws) {
    const int b = blockIdx.y;
    const float* Kb = ws + OFF_K + (size_t)b * NN * NN;
    const float* vb = ws + OFF_V + b * UV_STRIDE;
    float* ub       = ws + OFF_U + b * UV_STRIDE;
    const float kv = ws[OFF_KV];
    if (blockIdx.x < 128) {
        const int r = blockIdx.x * 8 + (threadIdx.x >> 5);
        const int lane = threadIdx.x & 31;
        const float4* row = (const float4*)(Kb + (size_t)r * NN);
        const float4* v4  = (const float4*)vb;
        float s = 0.f;
        #pragma unroll
        for (int it = 0; it < 8; ++it) {
            const int idx = it * 32 + lane;
            float4 kq = row[idx], vq = v4[idx];
            s += kq.x * vq.x + kq.y * vq.y + kq.z * vq.z + kq.w * vq.w;
        }
        #pragma unroll
        for (int off = 16; off > 0; off >>= 1) s += __shfl_down(s, off, 32);
        if (lane == 0) ub[r] = A_REAL / (s + kv * vb[NN] + EPS_);
    } else {
        __shared__ float red[256];
        float s = 0.f;
        for (int i = threadIdx.x; i < NN; i += 256) s += vb[i];
        red[threadIdx.x] = s; __syncthreads();
        for (int off = 128; off > 0; off >>= 1) {
            if ((int)threadIdx.x < off) red[threadIdx.x] += red[threadIdx.x + off];
            __syncthreads();
        }
        if (threadIdx.x == 0) ub[NN] = A_VIRT / (kv * red[0] + vb[NN] + EPS_);
    }
}

// 4b) transposed matvec, split-K over 8 row chunks for memory-level parallelism:
//     part[z][b][j] = sum_{i in chunk z} K[i][j] * u[i]   (column walk, coalesced)
__global__ __launch_bounds__(256)
void tmv_partial_kernel(float* ws) {
    const int b = blockIdx.y;
    const int z = blockIdx.z;
    const int j = blockIdx.x * 256 + threadIdx.x;
    const float* Kb = ws + OFF_K + (size_t)b * NN * NN + (size_t)z * CHUNK * NN;
    const float* ub = ws + OFF_U + b * UV_STRIDE + z * CHUNK;
    float s = 0.f;
    for (int i = 0; i < CHUNK; i += 4) {
        float4 uq = *(const float4*)(ub + i);
        const float* kp = Kb + (size_t)i * NN + j;
        s += kp[0] * uq.x + kp[NN] * uq.y + kp[2 * NN] * uq.z + kp[3 * NN] * uq.w;
    }
    ws[OFF_PART + ((size_t)z * BB + b) * NN + j] = s;
}

// 4c) finish v half-step: v_j = b_j / (Σ_z part + kv*u[N] + eps); block 4 does v[N]
__global__ __launch_bounds__(256)
void sink_cols_finish_kernel(float* ws) {
    const int b = blockIdx.y;
    const float* ub = ws + OFF_U + b * UV_STRIDE;
    float* vb       = ws + OFF_V + b * UV_STRIDE;
    const float kv = ws[OFF_KV];
    if (blockIdx.x < 4) {
        const int j = blockIdx.x * 256 + threadIdx.x;
        float s = 0.f;
        #pragma unroll
        for (int z = 0; z < NCHUNK; ++z) s += ws[OFF_PART + ((size_t)z * BB + b) * NN + j];
        vb[j] = A_REAL / (s + kv * ub[NN] + EPS_);
    } else {
        __shared__ float red[256];
        float s = 0.f;
        for (int i = threadIdx.x; i < NN; i += 256) s += ub[i];
        red[threadIdx.x] = s; __syncthreads();
        for (int off = 128; off > 0; off >>= 1) {
            if ((int)threadIdx.x < off) red[threadIdx.x] += red[threadIdx.x + off];
            __syncthreads();
        }
        if (threadIdx.x == 0) vb[NN] = A_VIRT / (kv * red[0] + ub[NN] + EPS_);
    }
}

// 5) real-only column sums for normalization: cs[b][x] = Σ_z part[z][b][x]
__global__ __launch_bounds__(256)
void colsum_finish_kernel(float* ws) {
    const int b = blockIdx.y;
    const int j = blockIdx.x * 256 + threadIdx.x;
    float s = 0.f;
    #pragma unroll
    for (int z = 0; z < NCHUNK; ++z) s += ws[OFF_PART + ((size_t)z * BB + b) * NN + j];
    ws[OFF_CS + b * NN + j] = s;
}

// 6) out[b][x][y] = u[y]*K[y][x] / cs[x]   (v_x cancels in the row-normalize)
//    32x32 LDS tile transpose; both global read and write fully coalesced.
__global__ __launch_bounds__(256)
void plan_out_kernel(const float* __restrict__ ws, float* __restrict__ out) {
    __shared__ float tile[32][33];
    const int b = blockIdx.z;
    const int x0 = blockIdx.x * 32, y0 = blockIdx.y * 32;
    const float* Kb = ws + OFF_K + (size_t)b * NN * NN;
    const float* ub = ws + OFF_U + b * UV_STRIDE;
    const float* cs = ws + OFF_CS + b * NN;
    const int tx = threadIdx.x;
    for (int yy = threadIdx.y; yy < 32; yy += 8) {
        const int y = y0 + yy;
        tile[yy][tx] = Kb[(size_t)y * NN + x0 + tx] * ub[y];
    }
    __syncthreads();
    float* ob = out + (size_t)b * NN * NN;
    for (int xx = threadIdx.y; xx < 32; xx += 8) {
        const int x = x0 + xx;
        ob[(size_t)x * NN + y0 + tx] = tile[tx][xx] / cs[x];
    }
}

// ---------------------------------------------------------------------------
extern "C" void kernel_launch(void* const* d_in, const int* in_sizes, int n_in,
                              void* d_out, int out_size, void* d_ws, size_t ws_size,
                              hipStream_t stream) {
    (void)in_sizes; (void)n_in; (void)out_size; (void)ws_size;
    const float* src = (const float*)d_in[0];
    const float* tgt = (const float*)d_in[1];
    float* ws  = (float*)d_ws;
    float* out = (float*)d_out;

    norms_kernel<<<dim3(4, 8, 2), 256, 0, stream>>>(src, tgt, ws);
    gemm_k_kernel<<<dim3(2, 64, 8), 256, 0, stream>>>(src, tgt, ws);

    // quantile 0.75 of 1,048,576 values -> order stats 786431 and 786432 (0-indexed)
    const unsigned ranks[2] = {786431u, 786432u};
    for (int sel = 0; sel < 2; ++sel) {
        radix_init_kernel<<<1, 256, 0, stream>>>(ws, ranks[sel]);
        for (int p = 0; p < 4; ++p) {
            const int shift = 24 - 8 * p;
            const unsigned mask = (p == 0) ? 0u : (0xFFFFFFFFu << (shift + 8));
            radix_count_kernel<<<256, 256, 0, stream>>>(ws, shift, mask);
            radix_scan_kernel<<<1, 32, 0, stream>>>(ws, shift, (p == 3) ? 1 : 0, sel);
        }
    }
    finalize_kv_kernel<<<1, 32, 0, stream>>>(ws);
    init_uv_kernel<<<8, 256, 0, stream>>>(ws);

    for (int it = 0; it < 50; ++it) {
        sink_rows_kernel<<<dim3(129, 8), 256, 0, stream>>>(ws);
        tmv_partial_kernel<<<dim3(4, 8, NCHUNK), 256, 0, stream>>>(ws);
        sink_cols_finish_kernel<<<dim3(5, 8), 256, 0, stream>>>(ws);
    }

    // final column sums (real rows only) reuse the split-K partial kernel
    tmv_partial_kernel<<<dim3(4, 8, NCHUNK), 256, 0, stream>>>(ws);
    colsum_finish_kernel<<<dim3(4, 8), 256, 0, stream>>>(ws);
    plan_out_kernel<<<dim3(32, 32, 8), dim3(32, 8), 0, stream>>>(ws, out);
}